// HybridAttentionMIL_15135464751637
// MI455X (gfx1250) — compile-verified
//
#include <hip/hip_runtime.h>
#include <math.h>

// ---- types for WMMA ----
typedef __attribute__((ext_vector_type(2))) float v2f;
typedef __attribute__((ext_vector_type(8))) float v8f;

#define EMBED 32
#define HID   16
#define NB1   1024   // blocks for kernel1
#define NB3   1024   // blocks for kernel3

// Raw v_rcp_f32 (1 ulp-ish) — fine here, softmax renormalizes.
__device__ __forceinline__ float fast_rcp(float x) {
    return __builtin_amdgcn_rcpf(x);
}
__device__ __forceinline__ float fast_sigmoid(float x) {
    x = fminf(fmaxf(x, -30.f), 30.f);
    return fast_rcp(1.0f + __expf(-x));
}
__device__ __forceinline__ float fast_tanh(float x) {
    x = fminf(fmaxf(x, -15.f), 15.f);
    float e = __expf(2.0f * x);
    return 1.0f - 2.0f * fast_rcp(e + 1.0f);
}

// -------------------------------------------------------------------------
// Kernel 1: a_raw[i] = (tanh(H Wv + bv) * sigmoid(H Wu + bu)) . Ww + bw
// One wave handles a 16-row tile via V_WMMA_F32_16X16X4_F32 (K=32 as 8x K4).
// A-matrix layout (16x4 f32): lanes 0-15 -> M=lane, VGPR0=K0,VGPR1=K1;
//                             lanes16-31 -> M=lane-16, VGPR0=K2,VGPR1=K3.
// B-matrix layout (4x16 f32): lanes hold N=lane&15; half selects K0/K2 rows.
// C layout: VGPR r = row (r + 8*half), column = lane&15.
// Bias folded into the accumulator init (same column for all 8 C elements).
// -------------------------------------------------------------------------
__global__ __launch_bounds__(256)
void k1_scores(const float* __restrict__ H,
               const float* __restrict__ Wv, const float* __restrict__ bv,
               const float* __restrict__ Wu, const float* __restrict__ bu,
               const float* __restrict__ Ww, const float* __restrict__ bw,
               float* __restrict__ a_out, float* __restrict__ bmax, int N)
{
    const int lane  = threadIdx.x & 31;
    const int wid   = threadIdx.x >> 5;
    const int half  = lane >> 4;           // 0: K0/K1 rows, 1: K2/K3 rows
    const int l16   = lane & 15;
    const int gwave = blockIdx.x * (blockDim.x >> 5) + wid;
    const int nwave = gridDim.x * (blockDim.x >> 5);

    // Preload B operands for both weight matrices (per-wave invariant).
    v2f bV[8], bU[8];
#pragma unroll
    for (int kk = 0; kk < 8; ++kk) {
        int k0 = 4 * kk + 2 * half;
        bV[kk].x = Wv[k0 * HID + l16];
        bV[kk].y = Wv[(k0 + 1) * HID + l16];
        bU[kk].x = Wu[k0 * HID + l16];
        bU[kk].y = Wu[(k0 + 1) * HID + l16];
    }
    const float bvv = bv[l16];
    const float buu = bu[l16];
    const float wl  = Ww[l16];
    const float bw0 = bw[0];

    const int ntiles = (N + 15) >> 4;
    float lmax = -INFINITY;

    for (int t = gwave; t < ntiles; t += nwave) {
        const int row0 = t * 16;
        int m = row0 + l16;
        if (m > N - 1) m = N - 1;                    // clamp (reads only)
        const float* hrow = H + (size_t)m * EMBED;

        v2f a[8];
#pragma unroll
        for (int kk = 0; kk < 8; ++kk) {
            int k0 = 4 * kk + 2 * half;
            a[kk] = *(const v2f*)(hrow + k0);        // 8B-aligned b64 load
        }

        // Bias-initialized accumulators (column bias uniform across rows).
        v8f cV, cU;
#pragma unroll
        for (int r = 0; r < 8; ++r) { cV[r] = bvv; cU[r] = buu; }
#pragma unroll
        for (int kk = 0; kk < 8; ++kk)
            cV = __builtin_amdgcn_wmma_f32_16x16x4_f32(false, a[kk], false, bV[kk],
                                                       (short)0, cV, false, false);
#pragma unroll
        for (int kk = 0; kk < 8; ++kk)
            cU = __builtin_amdgcn_wmma_f32_16x16x4_f32(false, a[kk], false, bU[kk],
                                                       (short)0, cU, false, false);

#pragma unroll
        for (int r = 0; r < 8; ++r) {
            float v = fast_tanh(cV[r]);
            float u = fast_sigmoid(cU[r]);
            float g = v * u * wl;                    // partial dot with Ww
            g += __shfl_xor(g, 1, 32);
            g += __shfl_xor(g, 2, 32);
            g += __shfl_xor(g, 4, 32);
            g += __shfl_xor(g, 8, 32);               // 16-lane group sum
            float araw = g + bw0;
            int row = row0 + r + 8 * half;
            if (l16 == r && row < N) a_out[row] = araw;
            if (row < N) lmax = fmaxf(lmax, araw);
        }
    }

    // block max
#pragma unroll
    for (int msk = 16; msk >= 1; msk >>= 1)
        lmax = fmaxf(lmax, __shfl_xor(lmax, msk, 32));
    __shared__ float smax[8];
    if (lane == 0) smax[wid] = lmax;
    __syncthreads();
    if (threadIdx.x == 0) {
        float mx = smax[0];
        for (int w = 1; w < 8; ++w) mx = fmaxf(mx, smax[w]);
        bmax[blockIdx.x] = mx;
    }
}

// ------------------------------------------------------------------
// Kernel 2: single block reduces per-block maxima -> global max
// ------------------------------------------------------------------
__global__ __launch_bounds__(256)
void k2_gmax(const float* __restrict__ bmax, int n, float* __restrict__ gmax)
{
    float m = -INFINITY;
    for (int i = threadIdx.x; i < n; i += blockDim.x) m = fmaxf(m, bmax[i]);
#pragma unroll
    for (int msk = 16; msk >= 1; msk >>= 1)
        m = fmaxf(m, __shfl_xor(m, msk, 32));
    __shared__ float s[8];
    int wid = threadIdx.x >> 5;
    if ((threadIdx.x & 31) == 0) s[wid] = m;
    __syncthreads();
    if (threadIdx.x == 0) {
        float mx = s[0];
        for (int w = 1; w < (int)(blockDim.x >> 5); ++w) mx = fmaxf(mx, s[w]);
        *gmax = mx;
    }
}

// ------------------------------------------------------------------
// Kernel 3: e[i] = exp(a[i]-gmax) (in place over a); per-block partial
// sums of S = sum(e) and Bcol[j] = sum(e_i * H[i][j]).  Lane == column.
// ------------------------------------------------------------------
__global__ __launch_bounds__(256)
void k3_expsum(const float* __restrict__ H, float* __restrict__ ae,
               const float* __restrict__ gmaxp, float* __restrict__ part, int N)
{
    const float gmax = *gmaxp;
    const int lane  = threadIdx.x & 31;
    const int wid   = threadIdx.x >> 5;
    const int gwave = blockIdx.x * (blockDim.x >> 5) + wid;
    const int nwave = gridDim.x * (blockDim.x >> 5);

    float accS = 0.0f;
    float accCol = 0.0f;                 // column = lane

    for (int base = gwave * 32; base < N; base += nwave * 32) {
        int i = base + lane;
        float ev = 0.0f;
        if (i < N) {
            ev = __expf(ae[i] - gmax);
            ae[i] = ev;                  // in-place: a_raw -> e
        }
        accS += ev;
#pragma unroll
        for (int r = 0; r < 32; ++r) {
            float evr = __shfl(ev, r, 32);
            int row = base + r; if (row > N - 1) row = N - 1;   // evr==0 if OOB
            accCol = fmaf(evr, H[(size_t)row * EMBED + lane], accCol);
        }
    }
    // wave-sum of S (Bcol stays per-lane: one column per lane)
#pragma unroll
    for (int msk = 16; msk >= 1; msk >>= 1)
        accS += __shfl_xor(accS, msk, 32);

    __shared__ float sS[8];
    __shared__ float sB[8][EMBED];
    if (lane == 0) sS[wid] = accS;
    sB[wid][lane] = accCol;
    __syncthreads();
    if (threadIdx.x < EMBED) {
        float s = 0.0f;
        for (int w = 0; w < 8; ++w) s += sB[w][threadIdx.x];
        part[(size_t)blockIdx.x * (EMBED + 1) + 1 + threadIdx.x] = s;
    }
    if (threadIdx.x == 0) {
        float s = 0.0f;
        for (int w = 0; w < 8; ++w) s += sS[w];
        part[(size_t)blockIdx.x * (EMBED + 1)] = s;
    }
}

// ------------------------------------------------------------------
// Kernel 4: final deterministic reduction + MLP head -> score, invS
// ------------------------------------------------------------------
__global__ __launch_bounds__(256)
void k4_head(const float* __restrict__ part, int nb,
             const float* __restrict__ TPL,
             const float* __restrict__ W1, const float* __restrict__ b1,
             const float* __restrict__ W2, const float* __restrict__ b2,
             float* __restrict__ invS_out, float* __restrict__ score_out)
{
    __shared__ float sf[EMBED + 1];
    __shared__ float sbf[EMBED + 1];
    __shared__ float sh[HID];
    const int t = threadIdx.x;
    if (t < EMBED + 1) {
        float s = 0.0f;
        for (int b = 0; b < nb; ++b) s += part[(size_t)b * (EMBED + 1) + t];
        sf[t] = s;
    }
    __syncthreads();
    const float invS = 1.0f / sf[0];
    if (t == 0) *invS_out = invS;
    if (t < EMBED) sbf[t] = sf[1 + t] * invS;   // B = (sum e_i H_i)/S
    if (t == EMBED) sbf[EMBED] = TPL[0];        // concat TPL
    __syncthreads();
    if (t < HID) {
        float acc = b1[t];
        for (int k = 0; k < EMBED + 1; ++k)
            acc = fmaf(sbf[k], W1[k * HID + t], acc);
        sh[t] = fmaxf(acc, 0.0f);               // relu
    }
    __syncthreads();
    if (t == 0) {
        float sc = b2[0];
        for (int n = 0; n < HID; ++n) sc += sh[n] * W2[n];
        *score_out = sc;
    }
}

// ------------------------------------------------------------------
// Kernel 5: alpha[i] = e[i] * invS -> d_out[1+i]
// ------------------------------------------------------------------
__global__ __launch_bounds__(256)
void k5_norm(const float* __restrict__ e, const float* __restrict__ invSp,
             float* __restrict__ alpha, int N)
{
    const float invS = *invSp;
    int i = blockIdx.x * blockDim.x + threadIdx.x;
    if (i < N) alpha[i] = e[i] * invS;
}

extern "C" void kernel_launch(void* const* d_in, const int* in_sizes, int n_in,
                              void* d_out, int out_size, void* d_ws, size_t ws_size,
                              hipStream_t stream) {
    const float* H   = (const float*)d_in[0];
    const float* TPL = (const float*)d_in[1];
    const float* Wv  = (const float*)d_in[2];
    const float* bv  = (const float*)d_in[3];
    const float* Wu  = (const float*)d_in[4];
    const float* bu  = (const float*)d_in[5];
    const float* Ww  = (const float*)d_in[6];
    const float* bw  = (const float*)d_in[7];
    const float* W1  = (const float*)d_in[8];
    const float* b1  = (const float*)d_in[9];
    const float* W2  = (const float*)d_in[10];
    const float* b2  = (const float*)d_in[11];

    const int N = in_sizes[0] / EMBED;

    float* ws    = (float*)d_ws;
    float* a_e   = ws;                    // N floats: a_raw, then e in place
    float* bmax  = a_e + N;               // NB1 floats
    float* gmax  = bmax + NB1;            // 1 float
    float* invS  = gmax + 1;              // 1 float
    float* part  = invS + 1;              // NB3 * 33 floats

    float* out   = (float*)d_out;         // out[0]=score, out[1..N]=alpha

    k1_scores<<<NB1, 256, 0, stream>>>(H, Wv, bv, Wu, bu, Ww, bw, a_e, bmax, N);
    k2_gmax  <<<1,   256, 0, stream>>>(bmax, NB1, gmax);
    k3_expsum<<<NB3, 256, 0, stream>>>(H, a_e, gmax, part, N);
    k4_head  <<<1,   256, 0, stream>>>(part, NB3, TPL, W1, b1, W2, b2, invS, out);
    k5_norm  <<<(N + 255) / 256, 256, 0, stream>>>(a_e, invS, out + 1, N);
}